// GraphCNNSequential_83932250898780
// MI455X (gfx1250) — compile-verified
//
#include <hip/hip_runtime.h>
#include <hip/hip_bf16.h>
#include <cstdint>

typedef float v2f __attribute__((ext_vector_type(2)));
typedef float v8f __attribute__((ext_vector_type(8)));

#define HID 64

// ---------------------------------------------------------------------------
// GEMM: out[r, 0:64] = (RELU ? relu(A[r,:]) : A[r,:]) @ W (+ bias)
// A: [nRows, K] row-major, W: [K, 64] row-major. nRows % 16 == 0, K % 4 == 0.
// One wave computes one 16x16 tile with V_WMMA_F32_16X16X4_F32.
// W staged once per block into LDS, pair-packed -> ds_load_2addr_stride64_b64.
// NTA=1 marks the A stream non-temporal (read-once; keep L2 for h/scatter).
// All offsets are 32-bit unsigned so vmem uses SGPR-base + 32-bit voffset.
// ---------------------------------------------------------------------------
template <int K, int RELU, int NTA>
__global__ void __launch_bounds__(256) gemm_k_to_64(const float* __restrict__ A,
                                                    const float* __restrict__ W,
                                                    const float* __restrict__ bias,
                                                    float* __restrict__ out,
                                                    int nRows) {
    __shared__ float Ws[K * HID];
    // Stage + repack: Ws[((k>>1)*64 + c)*2 + (k&1)] = W[k*64 + c]
    for (unsigned idx = threadIdx.x; idx < K * HID; idx += 256u) {
        const unsigned k = idx >> 6;
        const unsigned c = idx & 63u;
        Ws[(((k >> 1) << 6) + c) * 2u + (k & 1u)] = W[idx];
    }
    __syncthreads();

    const unsigned wib  = threadIdx.x >> 5;              // wave in block (0..7)
    const unsigned lane = threadIdx.x & 31u;
    const unsigned tm   = blockIdx.x * 2u + (wib >> 2);  // 2 row tiles / block
    const unsigned tn   = wib & 3u;                      // 4 col tiles (64/16)
    if (tm >= ((unsigned)nRows >> 4)) return;

    const unsigned half = lane >> 4;                     // ISA half-wave split
    const unsigned kb   = half * 2u;                     // K sub-offset
    const unsigned colBase = tn * 16u + (lane & 15u);

    const float* arow = A + (tm * 16u + (lane & 15u)) * (unsigned)K;
    const v2f*   Wp   = (const v2f*)Ws;                  // pair-packed B

    v8f acc = {};
    for (unsigned k = 0; k < (unsigned)K; k += 4u) {
        v2f av;
        if (NTA) av = __builtin_nontemporal_load((const v2f*)(arow + k + kb));
        else     av = *(const v2f*)(arow + k + kb);
        if (RELU) { av.x = fmaxf(av.x, 0.0f); av.y = fmaxf(av.y, 0.0f); }
        const v2f bv = Wp[(((k + kb) >> 1) << 6) + colBase];
        acc = __builtin_amdgcn_wmma_f32_16x16x4_f32(
            false, av, false, bv, (short)0, acc, false, false);
    }

    const float bb = bias ? bias[colBase] : 0.0f;
    const unsigned rowHalf = half * 8u;                  // D: M = r + 8*(lane>=16)
    #pragma unroll
    for (unsigned r = 0; r < 8u; ++r) {
        out[(tm * 16u + rowHalf + r) * (unsigned)HID + colBase] = acc[r] + bb;
    }
}

// ---------------------------------------------------------------------------
// Edge embedding: e = edge_attr @ W_edge + b_edge, scatter-added into h[src].
// Same WMMA/LDS structure (K = 32); edge_attr (205 MB, read-once) streams NT.
// Scatter uses 32-bit voffsets -> saddr-form global_atomic_add_f32.
// ---------------------------------------------------------------------------
__global__ void __launch_bounds__(256) edge_embed_scatter(
        const float* __restrict__ edge_attr,
        const float* __restrict__ W,
        const float* __restrict__ bias,
        const int* __restrict__ src,
        float* __restrict__ h,
        int E) {
    __shared__ float Ws[32 * HID];
    for (unsigned idx = threadIdx.x; idx < 32u * HID; idx += 256u) {
        const unsigned k = idx >> 6;
        const unsigned c = idx & 63u;
        Ws[(((k >> 1) << 6) + c) * 2u + (k & 1u)] = W[idx];
    }
    __syncthreads();

    const unsigned wib  = threadIdx.x >> 5;
    const unsigned lane = threadIdx.x & 31u;
    const unsigned tm   = blockIdx.x * 2u + (wib >> 2);
    const unsigned tn   = wib & 3u;
    if (tm >= ((unsigned)E >> 4)) return;

    const unsigned half = lane >> 4;
    const unsigned kb   = half * 2u;
    const unsigned colBase = tn * 16u + (lane & 15u);

    const float* arow = edge_attr + (tm * 16u + (lane & 15u)) * 32u;
    const v2f*   Wp   = (const v2f*)Ws;

    v8f acc = {};
    #pragma unroll
    for (unsigned k = 0; k < 32u; k += 4u) {
        v2f av = __builtin_nontemporal_load((const v2f*)(arow + k + kb));
        const v2f bv = Wp[(((k + kb) >> 1) << 6) + colBase];
        acc = __builtin_amdgcn_wmma_f32_16x16x4_f32(
            false, av, false, bv, (short)0, acc, false, false);
    }

    const float bb = bias[colBase];
    const unsigned rowHalf = half * 8u;
    #pragma unroll
    for (unsigned r = 0; r < 8u; ++r) {
        const unsigned edge = tm * 16u + rowHalf + r;
        const unsigned node = (unsigned)src[edge];
        atomicAdd(&h[node * (unsigned)HID + colBase], acc[r] + bb);
    }
}

// ---------------------------------------------------------------------------
// Degree / normalization
// ---------------------------------------------------------------------------
__global__ void deg_init(float* __restrict__ deg, int n) {
    unsigned i = blockIdx.x * blockDim.x + threadIdx.x;
    if (i < (unsigned)n) deg[i] = 1.0f;   // self-loop contribution
}

__global__ void deg_accum(const int* __restrict__ dst, float* __restrict__ deg, int E) {
    unsigned j = blockIdx.x * blockDim.x + threadIdx.x;
    if (j < (unsigned)E) atomicAdd(&deg[(unsigned)dst[j]], 1.0f);
}

__global__ void deg_to_dinv(float* __restrict__ deg, int n) {
    unsigned i = blockIdx.x * blockDim.x + threadIdx.x;
    if (i < (unsigned)n) deg[i] = rsqrtf(deg[i]);   // deg >= 1 (self-loops)
}

// ---------------------------------------------------------------------------
// GCN epilogue 1: out[i,c] = bias[c] + hw[i,c] * dinv[i]^2   (self-loop term)
// ---------------------------------------------------------------------------
__global__ void layer_init(const float* __restrict__ hw,
                           const float* __restrict__ dinv,
                           const float* __restrict__ bias,
                           float* __restrict__ out, int n) {
    unsigned idx = blockIdx.x * blockDim.x + threadIdx.x;
    if (idx >= (unsigned)n * HID) return;
    unsigned i = idx >> 6;
    unsigned c = idx & 63u;
    float d = dinv[i];
    out[idx] = bias[c] + hw[idx] * d * d;
}

// ---------------------------------------------------------------------------
// GCN epilogue 2: out[dst,:] += hw[src,:] * dinv[src]*dinv[dst]
// One wave per edge; scalar index loads, L2-absorbed f32 atomics, 32-bit offs.
// ---------------------------------------------------------------------------
__global__ void layer_scatter(const float* __restrict__ hw,
                              const float* __restrict__ dinv,
                              const int* __restrict__ src,
                              const int* __restrict__ dst,
                              float* __restrict__ out, int E) {
    const unsigned j = (blockIdx.x * blockDim.x + threadIdx.x) >> 5;
    const unsigned lane = threadIdx.x & 31u;
    if (j >= (unsigned)E) return;
    const unsigned s = (unsigned)src[j];
    const unsigned d = (unsigned)dst[j];
    const float w = dinv[s] * dinv[d];
    const unsigned sb = s * (unsigned)HID;
    const unsigned db = d * (unsigned)HID;
    atomicAdd(&out[db + lane],       hw[sb + lane]       * w);
    atomicAdd(&out[db + lane + 32u], hw[sb + lane + 32u] * w);
}

// ---------------------------------------------------------------------------
// Global mean pool accumulation (sums + counts via atomics)
// ---------------------------------------------------------------------------
__global__ void pool_accum(const float* __restrict__ h,
                           const int* __restrict__ batch,
                           float* __restrict__ gsum,
                           float* __restrict__ cnt, int n) {
    const unsigned i = (blockIdx.x * blockDim.x + threadIdx.x) >> 5;
    const unsigned lane = threadIdx.x & 31u;
    if (i >= (unsigned)n) return;
    const unsigned g = (unsigned)batch[i];
    const unsigned hb = i * (unsigned)HID;
    const unsigned gb = g * (unsigned)HID;
    atomicAdd(&gsum[gb + lane],       h[hb + lane]);
    atomicAdd(&gsum[gb + lane + 32u], h[hb + lane + 32u]);
    if (lane == 0u) atomicAdd(&cnt[g], 1.0f);
}

// ---------------------------------------------------------------------------
// Classifier: g = gsum/cnt; out = relu(g@Wc1+bc1)@Wc2 + bc2. One thread/graph.
// ---------------------------------------------------------------------------
__global__ void classifier(const float* __restrict__ gsum,
                           const float* __restrict__ cnt,
                           const float* __restrict__ Wc1,
                           const float* __restrict__ bc1,
                           const float* __restrict__ Wc2,
                           const float* __restrict__ bc2,
                           float* __restrict__ outp, int G) {
    unsigned g = blockIdx.x * blockDim.x + threadIdx.x;
    if (g >= (unsigned)G) return;
    float inv = 1.0f / fmaxf(cnt[g], 1.0f);
    float gm[HID];
    #pragma unroll
    for (unsigned i = 0; i < HID; ++i) gm[i] = gsum[g * (unsigned)HID + i] * inv;
    float acc = bc2[0];
    for (unsigned j = 0; j < 32u; ++j) {
        float t = bc1[j];
        #pragma unroll
        for (unsigned i = 0; i < HID; ++i) t += gm[i] * Wc1[i * 32u + j];
        t = fmaxf(t, 0.0f);
        acc += t * Wc2[j];
    }
    outp[g] = acc;
}

// ---------------------------------------------------------------------------
// Host launcher
// ---------------------------------------------------------------------------
extern "C" void kernel_launch(void* const* d_in, const int* in_sizes, int n_in,
                              void* d_out, int out_size, void* d_ws, size_t ws_size,
                              hipStream_t stream) {
    const float* x         = (const float*)d_in[0];
    const int*   edge_idx  = (const int*)d_in[1];
    const float* edge_attr = (const float*)d_in[2];
    const int*   batch     = (const int*)d_in[3];
    const float* W_node = (const float*)d_in[4];
    const float* b_node = (const float*)d_in[5];
    const float* W_edge = (const float*)d_in[6];
    const float* b_edge = (const float*)d_in[7];
    const float* W_g1 = (const float*)d_in[8];
    const float* b_g1 = (const float*)d_in[9];
    const float* W_g2 = (const float*)d_in[10];
    const float* b_g2 = (const float*)d_in[11];
    const float* W_g3 = (const float*)d_in[12];
    const float* b_g3 = (const float*)d_in[13];
    const float* W_c1 = (const float*)d_in[14];
    const float* b_c1 = (const float*)d_in[15];
    const float* W_c2 = (const float*)d_in[16];
    const float* b_c2 = (const float*)d_in[17];

    const int N = in_sizes[0] / 128;   // 100000
    const int E = in_sizes[2] / 32;    // 1600000
    const int G = out_size;            // 512
    const int* src = edge_idx;         // edge_index[0]
    const int* dst = edge_idx + E;     // edge_index[1]

    // Workspace carve-up (floats)
    float* h    = (float*)d_ws;                   // [N,64]
    float* hw   = h  + (size_t)N * HID;           // [N,64]
    float* dinv = hw + (size_t)N * HID;           // [N]
    float* gsum = dinv + N;                       // [G,64]
    float* cnt  = gsum + (size_t)G * HID;         // [G]

    const int TPB = 256;
    const int tilesN = N >> 4;                    // 6250
    const int tilesE = E >> 4;                    // 100000
    const int gemmBlocksN = (tilesN + 1) / 2;     // 2 row tiles / block
    const int gemmBlocksE = (tilesE + 1) / 2;

    // 1) node embedding: h = x @ W_node + b_node  (K = 128, NT stream of x)
    gemm_k_to_64<128, 0, 1><<<gemmBlocksN, TPB, 0, stream>>>(
        x, W_node, b_node, h, N);

    // 2) edge embedding scattered to source nodes (K = 32)
    edge_embed_scatter<<<gemmBlocksE, TPB, 0, stream>>>(
        edge_attr, W_edge, b_edge, src, h, E);

    // 3) symmetric GCN normalization
    deg_init<<<(N + TPB - 1) / TPB, TPB, 0, stream>>>(dinv, N);
    deg_accum<<<(E + TPB - 1) / TPB, TPB, 0, stream>>>(dst, dinv, E);
    deg_to_dinv<<<(N + TPB - 1) / TPB, TPB, 0, stream>>>(dinv, N);

    // 4) three GCNConv layers; ReLU fused into the next layer's GEMM read
    gemm_k_to_64<64, 0, 0><<<gemmBlocksN, TPB, 0, stream>>>(h, W_g1, nullptr, hw, N);
    layer_init<<<((long long)N * HID + TPB - 1) / TPB, TPB, 0, stream>>>(hw, dinv, b_g1, h, N);
    layer_scatter<<<((long long)E * 32 + TPB - 1) / TPB, TPB, 0, stream>>>(hw, dinv, src, dst, h, E);

    gemm_k_to_64<64, 1, 0><<<gemmBlocksN, TPB, 0, stream>>>(h, W_g2, nullptr, hw, N);
    layer_init<<<((long long)N * HID + TPB - 1) / TPB, TPB, 0, stream>>>(hw, dinv, b_g2, h, N);
    layer_scatter<<<((long long)E * 32 + TPB - 1) / TPB, TPB, 0, stream>>>(hw, dinv, src, dst, h, E);

    gemm_k_to_64<64, 1, 0><<<gemmBlocksN, TPB, 0, stream>>>(h, W_g3, nullptr, hw, N);
    layer_init<<<((long long)N * HID + TPB - 1) / TPB, TPB, 0, stream>>>(hw, dinv, b_g3, h, N);
    layer_scatter<<<((long long)E * 32 + TPB - 1) / TPB, TPB, 0, stream>>>(hw, dinv, src, dst, h, E);

    // 5) global mean pool
    hipMemsetAsync(gsum, 0, (size_t)G * HID * sizeof(float), stream);
    hipMemsetAsync(cnt, 0, (size_t)G * sizeof(float), stream);
    pool_accum<<<((long long)N * 32 + TPB - 1) / TPB, TPB, 0, stream>>>(h, batch, gsum, cnt, N);

    // 6) classifier
    classifier<<<(G + TPB - 1) / TPB, TPB, 0, stream>>>(
        gsum, cnt, W_c1, b_c1, W_c2, b_c2, (float*)d_out, G);
}